// DecoderSeqAdvanced_1649267442131
// MI455X (gfx1250) — compile-verified
//
#include <hip/hip_runtime.h>
#include <hip/hip_bf16.h>
#include <stdint.h>

// Problem constants (match reference)
#define Bc  64
#define Tc  2048
#define Hc  256
#define Dc  128
#define Vc  512
#define PHc 64
#define BT  (Bc * Tc)   // 131072 rows

typedef __attribute__((ext_vector_type(2))) float v2f;
typedef __attribute__((ext_vector_type(8))) float v8f;

// ---- CDNA5 async global->LDS path (guarded; sync fallback keeps compile OK) ----
#if __has_builtin(__builtin_amdgcn_global_load_async_to_lds_b128)
#define HAVE_ASYNC_LDS 1
// Builtin prototype (from toolchain diagnostic): (v4i AS1*, v4i AS3*, Imm i32, Imm i32)
typedef int v4i __attribute__((vector_size(16)));
typedef __attribute__((address_space(1))) v4i* gptr128;
typedef __attribute__((address_space(3))) v4i* lptr128;
// Build AS-qualified pointers via integer round-trip (generic->AS1 is value-
// preserving; generic LDS low 32 bits == LDS byte offset per ISA 10.2).
#define AS_GLOBAL128(p) ((gptr128)(uintptr_t)(p))
#define AS_LDS128(p)    ((lptr128)(unsigned)(uintptr_t)(p))
#if __has_builtin(__builtin_amdgcn_s_wait_asynccnt)
#define WAIT_ASYNC(n) __builtin_amdgcn_s_wait_asynccnt(n)
#else
#define WAIT_ASYNC(n) asm volatile("s_wait_asynccnt %0" ::"n"(n) : "memory")
#endif
#else
#define HAVE_ASYNC_LDS 0
#define WAIT_ASYNC(n)
#endif

// ---------------------------------------------------------------------------
// Kernel A: prefix offsets of (seq_len-1). offs[b] = sum_{i<b}(seq_len[i]-1)
// ---------------------------------------------------------------------------
__global__ void offs_kernel(const long long* __restrict__ seq_len, int* __restrict__ offs) {
    if (threadIdx.x == 0) {
        int acc = 0;
        for (int i = 0; i < Bc; ++i) {
            offs[i] = acc;
            acc += (int)seq_len[i] - 1;
        }
        offs[Bc] = acc; // == N
    }
}

// ---------------------------------------------------------------------------
// Kernel B: proj[m, d] = sum_h hidden[m,h] * fc_w[d,h] + fc_b[d]
// One wave computes a 16(row) x 128(D) tile via V_WMMA_F32_16X16X4_F32.
// ---------------------------------------------------------------------------
__global__ __launch_bounds__(256) void proj_kernel(
    const float* __restrict__ hidden, const float* __restrict__ fcw,
    const float* __restrict__ fcb, float* __restrict__ proj)
{
    const int lane = threadIdx.x & 31;
    const int wave = threadIdx.x >> 5;
    const int row0 = (blockIdx.x * 8 + wave) * 16;
    const int m    = lane & 15;
    const int kh   = (lane >> 4) * 2;   // K sub-pair select

    const float* Arow = hidden + (size_t)(row0 + m) * Hc + kh;

    v8f c[8];
#pragma unroll
    for (int ct = 0; ct < 8; ++ct) c[ct] = (v8f){};

#pragma unroll 4
    for (int k0 = 0; k0 < Hc; k0 += 4) {
        v2f a = *(const v2f*)(Arow + k0);
#pragma unroll
        for (int ct = 0; ct < 8; ++ct) {
            v2f b = *(const v2f*)(fcw + (size_t)(ct * 16 + m) * Hc + k0 + kh);
            c[ct] = __builtin_amdgcn_wmma_f32_16x16x4_f32(
                false, a, false, b, (short)0, c[ct], false, false);
        }
    }

    const int rbase = row0 + ((lane >> 4) << 3);
#pragma unroll
    for (int ct = 0; ct < 8; ++ct) {
        const int col  = ct * 16 + m;
        const float bv = fcb[col];
#pragma unroll
        for (int j = 0; j < 8; ++j) {
            proj[(size_t)(rbase + j) * Dc + col] = c[ct][j] + bv;
        }
    }
}

// ---------------------------------------------------------------------------
// Kernel C: per-row MLP as WMMA GEMM (l1_w staged in LDS).
// ---------------------------------------------------------------------------
__global__ __launch_bounds__(256) void mlp_kernel(
    const float* __restrict__ proj, const float* __restrict__ divv,
    const float* __restrict__ yrs,  const float* __restrict__ mask,
    const float* __restrict__ l1w,  const float* __restrict__ l1b,
    const float* __restrict__ l2w,  const float* __restrict__ l2b,
    float* __restrict__ wbuf)
{
    __shared__ __align__(16) float sW1[PHc * 130];   // [p][130]
    __shared__ float sb1[PHc];
    __shared__ float sW2[PHc];

    for (int i = threadIdx.x; i < PHc * 130; i += 256) sW1[i] = l1w[i];
    if (threadIdx.x < PHc) {
        sb1[threadIdx.x] = l1b[threadIdx.x];
        sW2[threadIdx.x] = l2w[threadIdx.x];
    }
    __syncthreads();

    const int lane = threadIdx.x & 31;
    const int wave = threadIdx.x >> 5;
    const int row0 = (blockIdx.x * 8 + wave) * 16;
    const int m    = lane & 15;
    const int kh   = (lane >> 4) * 2;

    const float* Arow = proj + (size_t)(row0 + m) * Dc + kh;

    v8f c[4];
#pragma unroll
    for (int ct = 0; ct < 4; ++ct) c[ct] = (v8f){};

#pragma unroll 4
    for (int k0 = 0; k0 < Dc; k0 += 4) {
        v2f a = *(const v2f*)(Arow + k0);
#pragma unroll
        for (int ct = 0; ct < 4; ++ct) {
            v2f b = *(const v2f*)(&sW1[(size_t)(ct * 16 + m) * 130 + 2 + k0 + kh]);
            c[ct] = __builtin_amdgcn_wmma_f32_16x16x4_f32(
                false, a, false, b, (short)0, c[ct], false, false);
        }
    }

    float w10[4], w11[4], b1v[4], w2v[4];
#pragma unroll
    for (int ct = 0; ct < 4; ++ct) {
        const int p = ct * 16 + m;
        w10[ct] = sW1[(size_t)p * 130 + 0];
        w11[ct] = sW1[(size_t)p * 130 + 1];
        b1v[ct] = sb1[p];
        w2v[ct] = sW2[p];
    }

    const int rbase = row0 + ((lane >> 4) << 3);
    float acc[8];
#pragma unroll
    for (int j = 0; j < 8; ++j) {
        const int row = rbase + j;
        const float dv = divv[row];
        const float yv = yrs[row];
        float s = 0.f;
#pragma unroll
        for (int ct = 0; ct < 4; ++ct) {
            float h = c[ct][j] + b1v[ct] + dv * w10[ct] + yv * w11[ct];
            h = fmaxf(h, 0.f);
            s += h * w2v[ct];
        }
        s += __shfl_xor(s, 1, 32);
        s += __shfl_xor(s, 2, 32);
        s += __shfl_xor(s, 4, 32);
        s += __shfl_xor(s, 8, 32);
        acc[j] = s;
    }

    const float b2 = l2b[0];
    if (m == 0) {
#pragma unroll
        for (int j = 0; j < 8; ++j) {
            const int row = rbase + j;
            const float ps = 1.f / (1.f + expf(-(acc[j] + b2)));
            const float mk = mask[row];
            wbuf[row] = (1.f - mk) * ps + mk * (1.f - ps);
        }
    }
}

// ---------------------------------------------------------------------------
// Kernel D: raw[n, v] = proj[srcRow(n), :] @ meta[v, :]   (ragged gather GEMM)
// Block = 8 n-tiles sharing ONE 256-wide v-half. B (meta) is double-buffered
// through LDS in 16-wide K slices via async global->LDS (ASYNCcnt) so all 8
// waves read identical ds_load_b64 fragments; meta traffic drops 8x.
// LDS rows padded to 20 floats: conflict-free ds_b64 + 16B-aligned b128 stage.
// ---------------------------------------------------------------------------
#define BSTRIDE 20
#define KCHUNK  16
#define NCHUNK  (Dc / KCHUNK)   // 8

__global__ __launch_bounds__(256) void raw_kernel(
    const float* __restrict__ proj, const float* __restrict__ meta,
    const int* __restrict__ offs, int N, float* __restrict__ raw)
{
    __shared__ __align__(16) float sB[2][256 * BSTRIDE]; // 2 x 20KB
    __shared__ int soffs[Bc + 1];
    if (threadIdx.x <= Bc) soffs[threadIdx.x] = offs[threadIdx.x];

    const int tid   = threadIdx.x;
    const int lane  = threadIdx.x & 31;
    const int wave  = threadIdx.x >> 5;
    const int group = blockIdx.x >> 1;        // group of 8 n-tiles
    const int vbase = (blockIdx.x & 1) * 256; // shared by whole block
    const int ntile = group * 8 + wave;
    const int m     = lane & 15;
    const int kh    = (lane >> 4) * 2;

    // Stage source for this thread: one 16-float row slice per chunk.
    const float* bsrc = meta + (size_t)(vbase + tid) * Dc;
    float*       bdst = &sB[0][0] + tid * BSTRIDE;

#if HAVE_ASYNC_LDS
#define STAGE16(S, D) __builtin_amdgcn_global_load_async_to_lds_b128(            \
        AS_GLOBAL128(S), AS_LDS128(D), 0, 0)
#else
#define STAGE16(S, D) (*(float4*)(D) = *(const float4*)(S))
#endif

#define STAGE(chunk, buf)                                                        \
    do {                                                                         \
        const float* _s = bsrc + (chunk) * KCHUNK;                               \
        float*       _d = bdst + (buf) * (256 * BSTRIDE);                        \
        _Pragma("unroll")                                                        \
        for (int _i = 0; _i < 4; ++_i) {                                         \
            STAGE16(_s + _i * 4, _d + _i * 4);                                   \
        }                                                                        \
    } while (0)

    // Gathered A row for this wave's n-tile (clamped; stores guarded).
    int n = ntile * 16 + m;
    if (n > N - 1) n = N - 1;
    if (n < 0)     n = 0;
    __syncthreads();                      // soffs visible
    int lo = 0, hi = Bc;                  // fixed 6-step search
    while (hi - lo > 1) {
        const int mid = (lo + hi) >> 1;
        if (soffs[mid] <= n) lo = mid; else hi = mid;
    }
    const int srcRow = lo * Tc + (n - soffs[lo]);
    const float* Arow = proj + (size_t)srcRow * Dc + kh;

    v8f c[16];
#pragma unroll
    for (int ct = 0; ct < 16; ++ct) c[ct] = (v8f){};

    STAGE(0, 0);                          // prologue: chunk 0 -> buf 0

    for (int ch = 0; ch < NCHUNK; ++ch) {
        const int cur = ch & 1;
        if (ch + 1 < NCHUNK) STAGE(ch + 1, 1 - cur);
#if HAVE_ASYNC_LDS
        if (ch + 1 < NCHUNK) { WAIT_ASYNC(4); } else { WAIT_ASYNC(0); }
#endif
        __syncthreads();                  // chunk ch staged by all waves

        const float* bbuf = &sB[cur][0];
#pragma unroll
        for (int kk = 0; kk < KCHUNK; kk += 4) {
            v2f a = *(const v2f*)(Arow + ch * KCHUNK + kk);
#pragma unroll
            for (int ct = 0; ct < 16; ++ct) {
                v2f b = *(const v2f*)(bbuf + (ct * 16 + m) * BSTRIDE + kk + kh);
                c[ct] = __builtin_amdgcn_wmma_f32_16x16x4_f32(
                    false, a, false, b, (short)0, c[ct], false, false);
            }
        }
        __syncthreads();                  // protect buf before next overwrite
    }

    const int nbase = ntile * 16 + ((lane >> 4) << 3);
#pragma unroll
    for (int ct = 0; ct < 16; ++ct) {
        const int v = vbase + ct * 16 + m;
#pragma unroll
        for (int j = 0; j < 8; ++j) {
            const int nn = nbase + j;
            if (nn < N) raw[(size_t)nn * Vc + v] = c[ct][j];
        }
    }
#undef STAGE
#undef STAGE16
}

// ---------------------------------------------------------------------------
// Kernel E: outlog[n] = log( w[srcRow(n)] )
// ---------------------------------------------------------------------------
__global__ __launch_bounds__(256) void logw_kernel(
    const float* __restrict__ wbuf, const int* __restrict__ offs,
    int N, float* __restrict__ outlog)
{
    const int n = blockIdx.x * 256 + threadIdx.x;
    if (n >= N) return;
    int lo = 0, hi = Bc;
    while (hi - lo > 1) {
        const int mid = (lo + hi) >> 1;
        if (offs[mid] <= n) lo = mid; else hi = mid;
    }
    const int srcRow = lo * Tc + (n - offs[lo]);
    outlog[n] = logf(wbuf[srcRow]);
}

// ---------------------------------------------------------------------------
// Launch
// ---------------------------------------------------------------------------
extern "C" void kernel_launch(void* const* d_in, const int* in_sizes, int n_in,
                              void* d_out, int out_size, void* d_ws, size_t ws_size,
                              hipStream_t stream) {
    const float*     hidden  = (const float*)d_in[0];
    const long long* seq_len = (const long long*)d_in[1];
    const float*     mask    = (const float*)d_in[2];
    const float*     divv    = (const float*)d_in[3];
    const float*     yrs     = (const float*)d_in[4];
    const float*     meta    = (const float*)d_in[5];
    const float*     fcw     = (const float*)d_in[6];
    const float*     fcb     = (const float*)d_in[7];
    const float*     l1w     = (const float*)d_in[8];
    const float*     l1b     = (const float*)d_in[9];
    const float*     l2w     = (const float*)d_in[10];
    const float*     l2b     = (const float*)d_in[11];

    // out = [raw (N*V floats)] ++ [log(cw) (N floats)]  =>  N = out_size / (V+1)
    const int N = out_size / (Vc + 1);

    int*   offs = (int*)d_ws;
    float* proj = (float*)((char*)d_ws + 512);
    float* wbuf = proj + (size_t)BT * Dc;

    float* raw    = (float*)d_out;
    float* outlog = raw + (size_t)N * Vc;

    offs_kernel<<<1, 32, 0, stream>>>(seq_len, offs);

    const int rowTileBlocks = (BT / 16) / 8;  // 1024 blocks, 8 waves * 16 rows each
    proj_kernel<<<rowTileBlocks, 256, 0, stream>>>(hidden, fcw, fcb, proj);
    mlp_kernel<<<rowTileBlocks, 256, 0, stream>>>(proj, divv, yrs, mask,
                                                  l1w, l1b, l2w, l2b, wbuf);

    const int ntiles  = (N + 15) / 16;
    const int ngroups = (ntiles + 7) / 8;      // 8 n-tiles per block
    raw_kernel<<<ngroups * 2, 256, 0, stream>>>(proj, meta, offs, N, raw);

    logw_kernel<<<(N + 255) / 256, 256, 0, stream>>>(wbuf, offs, N, outlog);
}